// GroupedQueryAttentionWithKVCache_15461882266177
// MI455X (gfx1250) — compile-verified
//
#include <hip/hip_runtime.h>
#include <math.h>

typedef __attribute__((ext_vector_type(2))) float v2f;
typedef __attribute__((ext_vector_type(8))) float v8f;

#define N_HEADS  32
#define KV_HEADS 8
#define HEAD_DIM 128
#define DIM      4096
#define KV_DIM   1024
#define QKV_N    (DIM + 2 * KV_DIM)   /* 6144 */
#define BATCH    32
#define CACHE_T  2048
#define KSPLIT   4                    /* K-dim split for memory-level parallelism */

// ---------------------------------------------------------------------------
// Partial C[M x N] = X[M x K-chunk] @ W^T  (W row-major [N x K]).
// One wave computes BOTH 16-row M-tiles of a 16-col N-tile, so each weight
// row is streamed exactly once (3 global_load_b64 per 2 WMMAs). K is split
// into KSPLIT chunks (blockIdx.y) writing partial tiles -> deterministic
// reduction pass afterwards (no float atomics).
// fp32 WMMA 16x16x4 frag layout: lanes 0-15 hold K={k,k+1}, lanes 16-31 hold
// K={k+2,k+3}; A lane = M-row, B lane = N-col -> per-lane contiguous float2.
// ---------------------------------------------------------------------------
__global__ __launch_bounds__(256)
void wmma_gemm_f32_ksplit(const float* __restrict__ X, int lda,
                          const float* __restrict__ W0,
                          const float* __restrict__ W1,
                          const float* __restrict__ W2,
                          int n1, int n2,
                          float* __restrict__ P, int ldc, int K, int Kc,
                          int Ntiles)
{
    const int lane  = threadIdx.x & 31;
    const int wid   = threadIdx.x >> 5;
    const int ntile = blockIdx.x * (blockDim.x >> 5) + wid;
    if (ntile >= Ntiles) return;
    const int n0   = ntile * 16;
    const int kbeg = blockIdx.y * Kc;

    const float* W; int nb;
    if (n0 < n1)      { W = W0; nb = n0; }
    else if (n0 < n2) { W = W1; nb = n0 - n1; }
    else              { W = W2; nb = n0 - n2; }

    const int   ksel = (lane >> 4) << 1;   // 0 (lanes 0-15) or 2 (lanes 16-31)
    const int   lm   = lane & 15;
    const float* A0 = X + (size_t)lm        * lda + kbeg + ksel;  // M rows 0-15
    const float* A1 = X + (size_t)(16 + lm) * lda + kbeg + ksel;  // M rows 16-31
    const float* B  = W + (size_t)(nb + lm) * K   + kbeg + ksel;  // N cols, streamed once

    v8f acc0 = {}, acc1 = {};
    for (int kb = 0; kb < Kc; kb += 32) {
        if (kb + 576 <= Kc) {              // prefetch 2KB ahead on each stream
            __builtin_prefetch(A0 + kb + 512, 0, 0);
            __builtin_prefetch(A1 + kb + 512, 0, 0);
            __builtin_prefetch(B  + kb + 512, 0, 0);
        }
        #pragma unroll
        for (int k = 0; k < 32; k += 4) {
            v2f a0 = *(const v2f*)(A0 + kb + k);
            v2f a1 = *(const v2f*)(A1 + kb + k);
            v2f b  = *(const v2f*)(B  + kb + k);
            acc0 = __builtin_amdgcn_wmma_f32_16x16x4_f32(
                       false, a0, false, b, (short)0, acc0, false, false);
            acc1 = __builtin_amdgcn_wmma_f32_16x16x4_f32(
                       false, a1, false, b, (short)0, acc1, false, false);
        }
    }

    // C/D layout: VGPR r -> M = r + 8*(lane>=16), N = lane & 15
    float* C0 = P + (size_t)blockIdx.y * BATCH * ldc
                  + (size_t)((lane >> 4) << 3) * ldc + (n0 + lm);
    #pragma unroll
    for (int r = 0; r < 8; ++r) {
        C0[(size_t)r * ldc]        = acc0[r];
        C0[(size_t)(r + 16) * ldc] = acc1[r];
    }
}

// ---------------------------------------------------------------------------
// Sum the KSPLIT partials of the QKV GEMM and apply RoPE in the same pass
// (q cols 0..4095, k cols 4096..5119; v passes through). One thread per
// (row, even-col pair) so the rotation is thread-local.
// ---------------------------------------------------------------------------
__global__ void reduce_rope_qkv(const float* __restrict__ P,
                                const float* __restrict__ fcos,
                                const float* __restrict__ fsin,
                                float* __restrict__ xqkv)
{
    const int half = QKV_N / 2;                     // 3072 pairs per row
    int idx = blockIdx.x * blockDim.x + threadIdx.x;
    if (idx >= BATCH * half) return;
    const int m = idx / half;
    const int c = (idx - m * half) * 2;
    const size_t off    = (size_t)m * QKV_N + c;
    const size_t stride = (size_t)BATCH * QKV_N;

    float e0 = P[off]     + P[off + stride]     + P[off + 2 * stride]     + P[off + 3 * stride];
    float e1 = P[off + 1] + P[off + 1 + stride] + P[off + 1 + 2 * stride] + P[off + 1 + 3 * stride];

    if (c < DIM + KV_DIM) {                        // q or k region -> RoPE
        const int   i  = (c & (HEAD_DIM - 1)) >> 1;
        const float co = fcos[i], si = fsin[i];
        const float r0 = e0 * co - e1 * si;
        const float r1 = e0 * si + e1 * co;
        e0 = r0; e1 = r1;
    }
    xqkv[off]     = e0;
    xqkv[off + 1] = e1;
}

// ---------------------------------------------------------------------------
// Sum the KSPLIT partials of the output projection into d_out.
// ---------------------------------------------------------------------------
__global__ void reduce_out(const float* __restrict__ P, float* __restrict__ out)
{
    int idx = blockIdx.x * blockDim.x + threadIdx.x;
    if (idx >= BATCH * DIM) return;
    const size_t stride = (size_t)BATCH * DIM;
    out[idx] = P[idx] + P[idx + stride] + P[idx + 2 * stride] + P[idx + 3 * stride];
}

// ---------------------------------------------------------------------------
// Attention for one (batch, kv_head): 4 rep-heads share each streamed K/V row.
// Keys 0..T-2 from cache, key T-1 is the fresh RoPE'd k (cache input is const).
// ---------------------------------------------------------------------------
__global__ __launch_bounds__(512)
void attn_kernel(const float* __restrict__ xqkv,
                 const float* __restrict__ cache_k,
                 const float* __restrict__ cache_v,
                 const int*   __restrict__ start_pos_p,
                 float* __restrict__ attn_out)
{
    __shared__ __align__(16) float sQ[4 * HEAD_DIM];
    __shared__ float sS[4 * CACHE_T];
    __shared__ float red[512];

    const int b     = blockIdx.x >> 3;
    const int kvh   = blockIdx.x & 7;
    const int tid   = threadIdx.x;
    const int T     = *start_pos_p + 1;
    const int tlast = T - 1;

    {
        const int r = tid >> 7, d = tid & 127;
        sQ[tid] = xqkv[(size_t)b * QKV_N + (kvh * 4 + r) * HEAD_DIM + d];
    }
    __syncthreads();

    const float scale = 0.08838834764831845f;      // 1/sqrt(128)
    const float* kfresh = xqkv + (size_t)b * QKV_N + DIM + kvh * HEAD_DIM;
    const float* vfresh = xqkv + (size_t)b * QKV_N + DIM + KV_DIM + kvh * HEAD_DIM;

    // scores: each thread owns key positions t, computes 4 rep-head dots
    for (int t = tid; t < T; t += 512) {
        const float* kp = (t == tlast) ? kfresh
            : cache_k + (((size_t)b * CACHE_T + t) * KV_HEADS + kvh) * HEAD_DIM;
        float a0 = 0.f, a1 = 0.f, a2 = 0.f, a3 = 0.f;
        #pragma unroll 8
        for (int d = 0; d < HEAD_DIM; d += 4) {
            const float4 kv = *(const float4*)(kp + d);
            const float4 q0 = *(const float4*)(sQ + 0 * HEAD_DIM + d);
            const float4 q1 = *(const float4*)(sQ + 1 * HEAD_DIM + d);
            const float4 q2 = *(const float4*)(sQ + 2 * HEAD_DIM + d);
            const float4 q3 = *(const float4*)(sQ + 3 * HEAD_DIM + d);
            a0 += kv.x * q0.x + kv.y * q0.y + kv.z * q0.z + kv.w * q0.w;
            a1 += kv.x * q1.x + kv.y * q1.y + kv.z * q1.z + kv.w * q1.w;
            a2 += kv.x * q2.x + kv.y * q2.y + kv.z * q2.z + kv.w * q2.w;
            a3 += kv.x * q3.x + kv.y * q3.y + kv.z * q3.z + kv.w * q3.w;
        }
        sS[0 * CACHE_T + t] = a0 * scale;
        sS[1 * CACHE_T + t] = a1 * scale;
        sS[2 * CACHE_T + t] = a2 * scale;
        sS[3 * CACHE_T + t] = a3 * scale;
    }
    __syncthreads();

    // softmax per rep-head r: 128 threads each
    const int r = tid >> 7, lane = tid & 127;
    float lm = -3.0e38f;
    for (int t = lane; t < T; t += 128) lm = fmaxf(lm, sS[r * CACHE_T + t]);
    red[tid] = lm;
    __syncthreads();
    for (int off = 64; off; off >>= 1) {
        if (lane < off) red[tid] = fmaxf(red[tid], red[tid + off]);
        __syncthreads();
    }
    const float rmax = red[r << 7];
    __syncthreads();

    float ls = 0.f;
    for (int t = lane; t < T; t += 128) {
        const float e = __expf(sS[r * CACHE_T + t] - rmax);
        sS[r * CACHE_T + t] = e;
        ls += e;
    }
    red[tid] = ls;
    __syncthreads();
    for (int off = 64; off; off >>= 1) {
        if (lane < off) red[tid] += red[tid + off];
        __syncthreads();
    }
    const float inv = 1.0f / red[r << 7];

    // P @ V: thread = (r, d); V rows coalesced across the 128 lanes.
    // 4 accumulators to break the serial FMA chain.
    const int d = lane;
    const float* vbase = cache_v + (((size_t)b * CACHE_T) * KV_HEADS + kvh) * HEAD_DIM + d;
    const size_t vstep = (size_t)KV_HEADS * HEAD_DIM;
    float acc0 = 0.f, acc1 = 0.f, acc2 = 0.f, acc3 = 0.f;
    int t = 0;
    for (; t + 4 <= tlast; t += 4) {
        acc0 += sS[r * CACHE_T + t + 0] * vbase[(size_t)(t + 0) * vstep];
        acc1 += sS[r * CACHE_T + t + 1] * vbase[(size_t)(t + 1) * vstep];
        acc2 += sS[r * CACHE_T + t + 2] * vbase[(size_t)(t + 2) * vstep];
        acc3 += sS[r * CACHE_T + t + 3] * vbase[(size_t)(t + 3) * vstep];
    }
    for (; t < tlast; ++t)
        acc0 += sS[r * CACHE_T + t] * vbase[(size_t)t * vstep];
    acc0 += sS[r * CACHE_T + tlast] * vfresh[d];

    attn_out[(size_t)b * DIM + (kvh * 4 + r) * HEAD_DIM + d] =
        (acc0 + acc1 + acc2 + acc3) * inv;
}

// ---------------------------------------------------------------------------
extern "C" void kernel_launch(void* const* d_in, const int* in_sizes, int n_in,
                              void* d_out, int out_size, void* d_ws, size_t ws_size,
                              hipStream_t stream)
{
    const float* x    = (const float*)d_in[0];
    const float* wq   = (const float*)d_in[1];
    const float* wk   = (const float*)d_in[2];
    const float* wv   = (const float*)d_in[3];
    const float* wo   = (const float*)d_in[4];
    const float* fcos = (const float*)d_in[5];
    const float* fsin = (const float*)d_in[6];
    const float* ck   = (const float*)d_in[7];
    const float* cv   = (const float*)d_in[8];
    const int*   sp   = (const int*)d_in[9];

    float* p_qkv    = (float*)d_ws;                          // 4 x 32 x 6144
    float* p_out    = p_qkv + (size_t)KSPLIT * BATCH * QKV_N; // 4 x 32 x 4096
    float* xqkv     = p_out + (size_t)KSPLIT * BATCH * DIM;   // 32 x 6144
    float* attn_out = xqkv  + (size_t)BATCH * QKV_N;          // 32 x 4096
    float* outp     = (float*)d_out;

    const int Kc = DIM / KSPLIT;   // 1024

    // 1) fused QKV projection, K-split partials
    {
        dim3 grid(QKV_N / 16 / 8, KSPLIT);
        wmma_gemm_f32_ksplit<<<grid, 256, 0, stream>>>(
            x, DIM, wq, wk, wv, DIM, DIM + KV_DIM,
            p_qkv, QKV_N, DIM, Kc, QKV_N / 16);
    }
    // 2) reduce partials + RoPE fused
    {
        const int total = BATCH * (QKV_N / 2);
        reduce_rope_qkv<<<(total + 255) / 256, 256, 0, stream>>>(
            p_qkv, fcos, fsin, xqkv);
    }
    // 3) attention (one workgroup per (batch, kv_head))
    attn_kernel<<<BATCH * KV_HEADS, 512, 0, stream>>>(xqkv, ck, cv, sp, attn_out);
    // 4) output projection, K-split partials
    {
        dim3 grid(DIM / 16 / 8, KSPLIT);
        wmma_gemm_f32_ksplit<<<grid, 256, 0, stream>>>(
            attn_out, DIM, wo, wo, wo, DIM, DIM,
            p_out, DIM, DIM, Kc, DIM / 16);
    }
    // 5) reduce output partials
    {
        const int total = BATCH * DIM;
        reduce_out<<<(total + 255) / 256, 256, 0, stream>>>(p_out, outp);
    }
}